// NEW_MINE_1279900254815
// MI455X (gfx1250) — compile-verified
//
#include <hip/hip_runtime.h>
#include <hip/hip_bf16.h>
#include <math.h>

#define NROWS  2048
#define AGENT  5
#define STATE  128
#define COMA   64
#define UNITS  32
#define LOG2F_ 0.6931471805599453f

typedef __attribute__((ext_vector_type(2))) float v2f;
typedef __attribute__((ext_vector_type(4))) float v4f;
typedef __attribute__((ext_vector_type(8))) float v8f;

// Fast, numerically stable softplus using hardware transcendentals
// (v_exp_f32 / v_log_f32). log argument is in [1,2] -> well conditioned;
// for large |x|, exp underflows to 0 and log(1)=0, which is exact.
__device__ __forceinline__ float softplus_f(float x) {
    return fmaxf(x, 0.0f) + __logf(1.0f + __expf(-fabsf(x)));
}

__device__ __forceinline__ float lrelu(float x) {
    return x > 0.0f ? x : 0.01f * x;
}

// ---------------------------------------------------------------------------
// Kernel 1: agent-pooled embeddings.
//   lbar[i][u] = 0.2 * sum_a lrelu( state[i,a,:] @ W1[:,u] + b1[u] )
//   mbar[i][u] = 0.2 * sum_a lrelu( joint_action[i,a,:] @ W2[:,u] + b2[u] )
// Block: 256 threads = 8 rows x 32 units. Grid: 256.
// ---------------------------------------------------------------------------
__global__ void embed_kernel(const float* __restrict__ state,
                             const float* __restrict__ action,
                             const float* __restrict__ W1,
                             const float* __restrict__ b1,
                             const float* __restrict__ W2,
                             const float* __restrict__ b2,
                             float* __restrict__ lbar,
                             float* __restrict__ mbar) {
    __shared__ float W1s[STATE * UNITS];   // 16 KB
    __shared__ float W2s[COMA * UNITS];    //  8 KB
    __shared__ float b1s[UNITS];
    __shared__ float b2s[UNITS];

    const int tid = threadIdx.x;
    for (int idx = tid; idx < STATE * UNITS; idx += 256) W1s[idx] = W1[idx];
    for (int idx = tid; idx < COMA  * UNITS; idx += 256) W2s[idx] = W2[idx];
    if (tid < UNITS) { b1s[tid] = b1[tid]; b2s[tid] = b2[tid]; }
    __syncthreads();

    const int u  = tid & 31;         // unit
    const int ly = tid >> 5;         // local row
    const int i  = blockIdx.x * 8 + ly;
    if (i >= NROWS) return;

    float sum1 = 0.0f;
    float sum2 = 0.0f;
    #pragma unroll
    for (int a = 0; a < AGENT; ++a) {
        const float* srow = state  + ((size_t)(i * AGENT + a)) * STATE;
        const float* arow = action + ((size_t)(i * AGENT + a)) * COMA;
        float acc1 = b1s[u];
        #pragma unroll 4
        for (int k = 0; k < STATE; ++k)
            acc1 = fmaf(srow[k], W1s[k * UNITS + u], acc1);
        float acc2 = b2s[u];
        #pragma unroll 4
        for (int k = 0; k < COMA; ++k)
            acc2 = fmaf(arow[k], W2s[k * UNITS + u], acc2);
        sum1 += lrelu(acc1);
        sum2 += lrelu(acc2);
    }
    lbar[i * UNITS + u] = sum1 * 0.2f;
    mbar[i * UNITS + u] = sum2 * 0.2f;
}

// ---------------------------------------------------------------------------
// Kernel 2: fused GEMM (ubar = mbar @ lbar^T) + softplus row-sum reduction.
// Uses V_WMMA_F32_16X16X4_F32 (native f32 WMMA, K=4, chained 8x for K=32).
//
// A layout (16x4 f32): lane L: M = L%16, h = L/16;
//   VGPR0 = A[M][kk+2h], VGPR1 = A[M][kk+2h+1]
// B layout (4x16 f32): lane L: N = L%16;
//   VGPR0 = B[kk+2h][N], VGPR1 = B[kk+2h+1][N]   (B[k][n] = lbar[j_base+n][k])
// C/D layout (16x16 f32): VGPR r: lanes 0-15 -> (M=r, N=lane),
//   lanes 16-31 -> (M=r+8, N=lane-16)
//
// Grid: 128 blocks (one per 16-row M band), 256 threads = 8 waves,
// wave w handles N tiles  bn = t*8 + w  for t = 0..15. Double-buffered B.
// No divergence in the tile loop (WMMA wants EXEC all-ones anyway).
// ---------------------------------------------------------------------------
#define NTILES_PER_WAVE 16
#define WAVES_PER_WG    8

__global__ void gemm_reduce_kernel(const float* __restrict__ mbar,
                                   const float* __restrict__ lbar,
                                   float* __restrict__ rowsum) {
    __shared__ float red[16];
    if (threadIdx.x < 16) red[threadIdx.x] = 0.0f;
    __syncthreads();

    const int lane = threadIdx.x & 31;
    const int w    = threadIdx.x >> 5;       // wave id 0..7
    const int h    = lane >> 4;              // half-wave select
    const int l15  = lane & 15;
    const int i_base = blockIdx.x * 16;
    const int koff   = 2 * h;                // per-lane K offset within K4 block

    // A fragment: mbar rows i_base..i_base+15, all K=0..31 (invariant over N)
    const float* mrow = mbar + (size_t)(i_base + l15) * UNITS;
    v2f afr[8];
    #pragma unroll
    for (int r = 0; r < 8; ++r)
        afr[r] = *(const v2f*)(mrow + 4 * r + koff);

    float acc[8];
    #pragma unroll
    for (int r = 0; r < 8; ++r) acc[r] = 0.0f;

    // prologue: load B fragments for first tile
    const float* lrow0 = lbar + (size_t)(w * 16 + l15) * UNITS;
    v2f bfr[8];
    #pragma unroll
    for (int r = 0; r < 8; ++r)
        bfr[r] = *(const v2f*)(lrow0 + 4 * r + koff);

    for (int t = 0; t < NTILES_PER_WAVE; ++t) {
        // prefetch next tile's B fragments while WMMA + softplus run
        v2f bnx[8];
        if (t + 1 < NTILES_PER_WAVE) {
            const int bn_next = (t + 1) * WAVES_PER_WG + w;
            const float* lrow = lbar + (size_t)(bn_next * 16 + l15) * UNITS;
            #pragma unroll
            for (int r = 0; r < 8; ++r)
                bnx[r] = *(const v2f*)(lrow + 4 * r + koff);
        }

        v8f c = {};
        #pragma unroll
        for (int r = 0; r < 8; ++r) {
            c = __builtin_amdgcn_wmma_f32_16x16x4_f32(
                    /*neg_a=*/false, afr[r],
                    /*neg_b=*/false, bfr[r],
                    /*c_mod=*/(short)0, c,
                    /*reuse_a=*/false, /*reuse_b=*/false);
        }

        #pragma unroll
        for (int r = 0; r < 8; ++r)
            acc[r] += softplus_f(c[r]);

        #pragma unroll
        for (int r = 0; r < 8; ++r)
            bfr[r] = bnx[r];
    }

    // reduce acc[r] across each 16-lane half (rows m=r for h=0, m=r+8 for h=1)
    #pragma unroll
    for (int r = 0; r < 8; ++r) {
        float v = acc[r];
        v += __shfl_xor(v, 8, 16);
        v += __shfl_xor(v, 4, 16);
        v += __shfl_xor(v, 2, 16);
        v += __shfl_xor(v, 1, 16);
        if (l15 == 0)
            atomicAdd(&red[r + 8 * h], v);   // ds_add_f32
    }
    __syncthreads();
    if (threadIdx.x < 16)
        rowsum[i_base + threadIdx.x] = red[threadIdx.x];
}

// ---------------------------------------------------------------------------
// Kernel 3: finalize loss / MI per row. Recomputes the diagonal
//   u_ii = mbar[i,:] . lbar[i,:]   (32-length dot, trivial)
// which keeps the GEMM kernel branch-free.
//   S       = sum_j softplus(u[i,j])
//   E_neg_t = (S - softplus(u_ii))/(n-1) - log2
//   E_pos_t = log2 - softplus(-u_ii)
//   loss = E_neg_t - E_pos_t ; MI = E_pos_t
// ---------------------------------------------------------------------------
__global__ void finalize_kernel(const float* __restrict__ rowsum,
                                const float* __restrict__ mbar,
                                const float* __restrict__ lbar,
                                float* __restrict__ out) {
    const int i = blockIdx.x * blockDim.x + threadIdx.x;
    if (i >= NROWS) return;

    const v4f* mrow = (const v4f*)(mbar + (size_t)i * UNITS);
    const v4f* lrow = (const v4f*)(lbar + (size_t)i * UNITS);
    float uii = 0.0f;
    #pragma unroll
    for (int q = 0; q < UNITS / 4; ++q) {
        v4f mv = mrow[q];
        v4f lv = lrow[q];
        uii = fmaf(mv.x, lv.x, uii);
        uii = fmaf(mv.y, lv.y, uii);
        uii = fmaf(mv.z, lv.z, uii);
        uii = fmaf(mv.w, lv.w, uii);
    }

    const float S   = rowsum[i];
    const float inv = 1.0f / (float)(NROWS - 1);
    const float E_neg_term = (S - softplus_f(uii)) * inv - LOG2F_;
    const float E_pos_term = LOG2F_ - softplus_f(-uii);
    out[i]         = E_neg_term - E_pos_term;  // loss
    out[NROWS + i] = E_pos_term;               // MI
}

extern "C" void kernel_launch(void* const* d_in, const int* in_sizes, int n_in,
                              void* d_out, int out_size, void* d_ws, size_t ws_size,
                              hipStream_t stream) {
    const float* state  = (const float*)d_in[0];
    const float* action = (const float*)d_in[1];
    const float* W1     = (const float*)d_in[2];
    const float* b1     = (const float*)d_in[3];
    const float* W2     = (const float*)d_in[4];
    const float* b2     = (const float*)d_in[5];
    float* out = (float*)d_out;

    float* ws     = (float*)d_ws;
    float* lbar   = ws;                         // 2048*32
    float* mbar   = ws + NROWS * UNITS;         // 2048*32
    float* rowsum = ws + 2 * NROWS * UNITS;     // 2048

    embed_kernel<<<NROWS / 8, 256, 0, stream>>>(state, action, W1, b1, W2, b2,
                                                lbar, mbar);
    gemm_reduce_kernel<<<NROWS / 16, WAVES_PER_WG * 32, 0, stream>>>(mbar, lbar,
                                                                     rowsum);
    finalize_kernel<<<NROWS / 256, 256, 0, stream>>>(rowsum, mbar, lbar, out);
}